// RNABasePairPredictor_82471962018256
// MI455X (gfx1250) — compile-verified
//
#include <hip/hip_runtime.h>
#include <hip/hip_bf16.h>
#include <math.h>

typedef __attribute__((ext_vector_type(2))) float v2f;
typedef __attribute__((ext_vector_type(8))) float v8f;

// ---------------------------------------------------------------------------
// Conv1: x (4,512,4) -> relu(conv k=3 pad=1) -> out (4,512,32)
// ---------------------------------------------------------------------------
__global__ void conv1_kernel(const float* __restrict__ x, const float* __restrict__ w,
                             const float* __restrict__ bias, float* __restrict__ out)
{
    int idx = blockIdx.x * blockDim.x + threadIdx.x;   // 4*512*32 = 65536
    int c = idx & 31;
    int l = (idx >> 5) & 511;
    int b = idx >> 14;
    float s = bias[c];
#pragma unroll
    for (int t = 0; t < 3; ++t) {
        int lp = l + t - 1;
        if ((unsigned)lp < 512u) {
            const float* xp = x + ((size_t)(b * 512 + lp) << 2);
            const float* wp = w + c * 12 + t;          // w[c][ci][t], stride 3 over ci
            s += xp[0] * wp[0] + xp[1] * wp[3] + xp[2] * wp[6] + xp[3] * wp[9];
        }
    }
    out[idx] = fmaxf(s, 0.f);                          // idx == (b*512+l)*32 + c
}

// ---------------------------------------------------------------------------
// Conv2: in (4,512,32) -> relu(conv k=3 pad=1) -> h (4,512,64)
// ---------------------------------------------------------------------------
__global__ void conv2_kernel(const float* __restrict__ in, const float* __restrict__ w,
                             const float* __restrict__ bias, float* __restrict__ out)
{
    int idx = blockIdx.x * blockDim.x + threadIdx.x;   // 4*512*64 = 131072
    int c2 = idx & 63;
    int l = (idx >> 6) & 511;
    int b = idx >> 15;
    float s = bias[c2];
#pragma unroll
    for (int t = 0; t < 3; ++t) {
        int lp = l + t - 1;
        if ((unsigned)lp < 512u) {
            const float* hp = in + (size_t)(b * 512 + lp) * 32;
            const float* wp = w + c2 * 96 + t;         // w[c2][c1][t], stride 3 over c1
#pragma unroll
            for (int c1 = 0; c1 < 32; ++c1) s += hp[c1] * wp[c1 * 3];
        }
    }
    out[idx] = fmaxf(s, 0.f);                          // idx == (b*512+l)*64 + c2
}

// ---------------------------------------------------------------------------
// WMMA f32 GEMM:  C[m][n] = sum_k A[m*K+k] * W[n*ldw+k]  (+ bias1[n] + bias2[n])
// One wave per 16x16 tile of C; K stepped by 4 with V_WMMA_F32_16X16X4_F32.
// Fragment layouts per CDNA5 ISA 7.12.2:
//   A 16x4:  lane<16 holds row M=l16, v0=K+0 v1=K+1 ; lane>=16: v0=K+2 v1=K+3
//   B 4x16:  N = lane%16 ; lane<16: v0=row K+0 v1=row K+1 ; lane>=16: K+2/K+3
//   C/D:     VGPR r: lanes0-15 -> M=r, lanes16-31 -> M=r+8, N=lane%16
// ---------------------------------------------------------------------------
__global__ void wmma_gemm_nt(const float* __restrict__ A, const float* __restrict__ W,
                             const float* __restrict__ bias1, const float* __restrict__ bias2,
                             float* __restrict__ C, int M, int N, int K, int ldw)
{
    const int lane = threadIdx.x & 31;
    const int wid  = (blockIdx.x * blockDim.x + threadIdx.x) >> 5;
    const int tiles_n = N >> 4;
    const int tm = wid / tiles_n;
    const int tn = wid - tm * tiles_n;
    const int row0 = tm << 4, col0 = tn << 4;
    const int half = lane >> 4;     // 0: K+0/K+1, 1: K+2/K+3
    const int l16  = lane & 15;

    const float* arow = A + (size_t)(row0 + l16) * K   + (half << 1);
    const float* wrow = W + (size_t)(col0 + l16) * ldw + (half << 1);

    v8f acc = {0.f, 0.f, 0.f, 0.f, 0.f, 0.f, 0.f, 0.f};
    for (int k = 0; k < K; k += 4) {
        v2f af, bf;
        af[0] = arow[k]; af[1] = arow[k + 1];
        bf[0] = wrow[k]; bf[1] = wrow[k + 1];
        acc = __builtin_amdgcn_wmma_f32_16x16x4_f32(false, af, false, bf,
                                                    (short)0, acc, false, false);
    }
    float bias = 0.f;
    if (bias1) bias += bias1[col0 + l16];
    if (bias2) bias += bias2[col0 + l16];
#pragma unroll
    for (int r = 0; r < 8; ++r)
        C[(size_t)(row0 + r + half * 8) * N + col0 + l16] = acc[r] + bias;
}

// ---------------------------------------------------------------------------
// Persistent LSTM recurrence. grid=2 (dir 0 fwd, dir 1 bwd), 256 threads.
// Thread g owns gate row g: keeps w_hh[g][0:64] in registers.
// h state (4x64) and gate pre-activations (4x256) live in LDS.
// Thread g also owns cell (b=g/64, hc=g%64): c kept in a register.
// ---------------------------------------------------------------------------
__global__ void lstm_kernel(const float* __restrict__ xg_f, const float* __restrict__ xg_b,
                            const float* __restrict__ whh_f, const float* __restrict__ whh_b,
                            float* __restrict__ hf, float* __restrict__ hb)
{
    const int dir = blockIdx.x;
    const float* xg  = dir ? xg_b  : xg_f;
    const float* whh = dir ? whh_b : whh_f;
    float*       out = dir ? hb    : hf;

    const int g = threadIdx.x;          // 0..255 : gate column
    __shared__ float h_s[4][64];
    __shared__ float gate_s[4][256];

    float wrow[64];
#pragma unroll
    for (int k = 0; k < 64; ++k) wrow[k] = whh[g * 64 + k];

    const int bo = g >> 6;              // owned batch
    const int hc = g & 63;              // owned hidden column
    float c = 0.f;
    h_s[bo][hc] = 0.f;
    __syncthreads();

    for (int step = 0; step < 512; ++step) {
        const int t = dir ? (511 - step) : step;
#pragma unroll
        for (int b = 0; b < 4; ++b) {
            float s = xg[(size_t)(b * 512 + t) * 256 + g];
#pragma unroll
            for (int k = 0; k < 64; ++k) s += wrow[k] * h_s[b][k];
            gate_s[b][g] = s;
        }
        __syncthreads();                 // gates ready, old h reads done
        float gi = gate_s[bo][hc];
        float gf = gate_s[bo][64 + hc];
        float gg = gate_s[bo][128 + hc];
        float go = gate_s[bo][192 + hc];
        float si = 1.f / (1.f + expf(-gi));
        float sf = 1.f / (1.f + expf(-gf));
        float so = 1.f / (1.f + expf(-go));
        c = sf * c + si * tanhf(gg);
        float hv = so * tanhf(c);
        h_s[bo][hc] = hv;
        out[(size_t)(bo * 512 + t) * 64 + hc] = hv;
        __syncthreads();                 // h updated before next step reads
    }
}

// ---------------------------------------------------------------------------
// Concat hf|hb + LayerNorm(128). One block of 128 threads per (b,l) row.
// ---------------------------------------------------------------------------
__global__ void ln_kernel(const float* __restrict__ hf, const float* __restrict__ hb,
                          const float* __restrict__ g, const float* __restrict__ bta,
                          float* __restrict__ z)
{
    const int m = blockIdx.x;           // 0..2047 == b*512+l
    const int d = threadIdx.x;          // 0..127
    float v = (d < 64) ? hf[(size_t)m * 64 + d] : hb[(size_t)m * 64 + (d - 64)];

    __shared__ float red[128];
    red[d] = v;
    __syncthreads();
    for (int s = 64; s > 0; s >>= 1) { if (d < s) red[d] += red[d + s]; __syncthreads(); }
    float mu = red[0] * (1.f / 128.f);
    __syncthreads();
    red[d] = v * v;
    __syncthreads();
    for (int s = 64; s > 0; s >>= 1) { if (d < s) red[d] += red[d + s]; __syncthreads(); }
    float var = red[0] * (1.f / 128.f) - mu * mu;
    z[(size_t)m * 128 + d] = (v - mu) * rsqrtf(var + 1e-5f) * g[d] + bta[d];
}

// ---------------------------------------------------------------------------
// Pairwise score: out[b,i,j] = sum_o relu(a[b,i,o]+b2[b,j,o]+pw_b[o])*sc_w[o] + sc_b
// 32x32 (i,j) tile per block; a/b2 rows staged in LDS as float4.
// Wave = 32 consecutive j at a common i (a read is a broadcast).
// ---------------------------------------------------------------------------
__global__ void pair_score_kernel(const float* __restrict__ A, const float* __restrict__ B2,
                                  const float* __restrict__ pwb, const float* __restrict__ scw,
                                  const float* __restrict__ scb, float* __restrict__ out)
{
    __shared__ float4 a_s[32][33];
    __shared__ float4 b_s[32][33];
    __shared__ float4 pb_s[32];
    __shared__ float4 sw_s[32];

    const int tid = threadIdx.x;        // 256 threads
    const int b  = blockIdx.z;
    const int i0 = blockIdx.y << 5;
    const int j0 = blockIdx.x << 5;

    const float4* Ag = (const float4*)(A  + (size_t)(b * 512 + i0) * 128);
    const float4* Bg = (const float4*)(B2 + (size_t)(b * 512 + j0) * 128);
    for (int idx = tid; idx < 1024; idx += 256) {
        int r = idx >> 5, cc = idx & 31;
        a_s[r][cc] = Ag[r * 32 + cc];
        b_s[r][cc] = Bg[r * 32 + cc];
    }
    if (tid < 32) { pb_s[tid] = ((const float4*)pwb)[tid]; sw_s[tid] = ((const float4*)scw)[tid]; }
    __syncthreads();

    const int j = tid & 31;
    const int ib = tid >> 5;            // 0..7
    const float sb = scb[0];
#pragma unroll
    for (int ii = 0; ii < 4; ++ii) {
        int i = ib + (ii << 3);
        float s = 0.f;
#pragma unroll 4
        for (int o = 0; o < 32; ++o) {
            float4 av = a_s[i][o], bv = b_s[j][o], pb = pb_s[o], sw = sw_s[o];
            s += fmaxf(av.x + bv.x + pb.x, 0.f) * sw.x;
            s += fmaxf(av.y + bv.y + pb.y, 0.f) * sw.y;
            s += fmaxf(av.z + bv.z + pb.z, 0.f) * sw.z;
            s += fmaxf(av.w + bv.w + pb.w, 0.f) * sw.w;
        }
        out[((size_t)b << 18) + (size_t)(i0 + i) * 512 + (j0 + j)] = s + sb;
    }
}

// ---------------------------------------------------------------------------
extern "C" void kernel_launch(void* const* d_in, const int* in_sizes, int n_in,
                              void* d_out, int out_size, void* d_ws, size_t ws_size,
                              hipStream_t stream)
{
    (void)in_sizes; (void)n_in; (void)out_size; (void)ws_size;
    const float* x     = (const float*)d_in[0];
    /* d_in[1] = lengths : unused by the reference */
    const float* c1w   = (const float*)d_in[2];
    const float* c1b   = (const float*)d_in[3];
    const float* c2w   = (const float*)d_in[4];
    const float* c2b   = (const float*)d_in[5];
    const float* wih_f = (const float*)d_in[6];
    const float* whh_f = (const float*)d_in[7];
    const float* bih_f = (const float*)d_in[8];
    const float* bhh_f = (const float*)d_in[9];
    const float* wih_b = (const float*)d_in[10];
    const float* whh_b = (const float*)d_in[11];
    const float* bih_b = (const float*)d_in[12];
    const float* bhh_b = (const float*)d_in[13];
    const float* ln_g  = (const float*)d_in[14];
    const float* ln_b  = (const float*)d_in[15];
    const float* pw_w  = (const float*)d_in[16];
    const float* pw_b  = (const float*)d_in[17];
    const float* sc_w  = (const float*)d_in[18];
    const float* sc_b  = (const float*)d_in[19];
    float* outp = (float*)d_out;

    float* ws     = (float*)d_ws;
    float* conv1o = ws;                   // 4*512*32  = 65536
    float* h      = conv1o + 65536;       // 4*512*64  = 131072
    float* xg_f   = h      + 131072;      // 4*512*256 = 524288
    float* xg_b   = xg_f   + 524288;      // 524288
    float* hf     = xg_b   + 524288;      // 131072
    float* hb     = hf     + 131072;      // 131072
    float* z      = hb     + 131072;      // 4*512*128 = 262144
    float* a_     = z      + 262144;      // 262144
    float* b2_    = a_     + 262144;      // 262144   (total ~8.75 MB)

    conv1_kernel<<<256, 256, 0, stream>>>(x, c1w, c1b, conv1o);
    conv2_kernel<<<512, 256, 0, stream>>>(conv1o, c2w, c2b, h);

    // xg = h @ w_ih^T + b_ih + b_hh  : (2048x64)@(64x256), 2048 tiles -> 512 blocks x 4 waves
    wmma_gemm_nt<<<512, 128, 0, stream>>>(h, wih_f, bih_f, bhh_f, xg_f, 2048, 256, 64, 64);
    wmma_gemm_nt<<<512, 128, 0, stream>>>(h, wih_b, bih_b, bhh_b, xg_b, 2048, 256, 64, 64);

    lstm_kernel<<<2, 256, 0, stream>>>(xg_f, xg_b, whh_f, whh_b, hf, hb);
    ln_kernel<<<2048, 128, 0, stream>>>(hf, hb, ln_g, ln_b, z);

    // a = z @ wi^T, b2 = z @ wj^T : (2048x128)@(128x128), W rows strided by 256
    wmma_gemm_nt<<<256, 128, 0, stream>>>(z, pw_w,       nullptr, nullptr, a_,  2048, 128, 128, 256);
    wmma_gemm_nt<<<256, 128, 0, stream>>>(z, pw_w + 128, nullptr, nullptr, b2_, 2048, 128, 128, 256);

    pair_score_kernel<<<dim3(16, 16, 4), 256, 0, stream>>>(a_, b2_, pw_b, sc_w, sc_b, outp);
}